// Motion_Loss_Module_29472065585163
// MI455X (gfx1250) — compile-verified
//
#include <hip/hip_runtime.h>
#include <hip/hip_bf16.h>

// ---------------------------------------------------------------------------
// Motion loss (scene-flow Kabsch loss) for MI455X / gfx1250.
// Heavy kernel: brute-force NN match via V_WMMA_F32_16X16X4_F32 (K=3 padded
// to 4), LDS-staged target tiles, wave32 xor-shuffle argmin reduction.
// ---------------------------------------------------------------------------

#define Bc   4
#define Nc   8192
#define Hc   256
#define Wc   256
#define Zc   20
#define SPc  60
#define ITERc 2
#define CHUNK 2048         // target points staged in LDS per pass (24 KB)

typedef float v2f __attribute__((ext_vector_type(2)));
typedef float v8f __attribute__((ext_vector_type(8)));

// ---------------------------------------------------------------- utilities
static __device__ __forceinline__ float det3(const float M[3][3]) {
    return M[0][0]*(M[1][1]*M[2][2]-M[1][2]*M[2][1])
         - M[0][1]*(M[1][0]*M[2][2]-M[1][2]*M[2][0])
         + M[0][2]*(M[1][0]*M[2][1]-M[1][1]*M[2][0]);
}

// ------------------------------------------------------------ init / zeroing
__global__ void ml_init_kernel(float* fgmap, float* counts, float* scal) {
    int idx = blockIdx.x * blockDim.x + threadIdx.x;
    int total = Bc * Hc * Wc;
    if (idx < total) fgmap[idx] = 0.0f;
    if (idx < Bc * SPc) counts[idx] = 0.0f;
    if (idx < Bc * 8) scal[idx] = 0.0f;
}

__global__ void ml_zero_seg_kernel(float* seg) {
    int idx = blockIdx.x * blockDim.x + threadIdx.x;
    for (int i = idx; i < Bc * SPc * 16; i += blockDim.x * gridDim.x) seg[i] = 0.0f;
}

// --------------------------------------------------- per-point preprocessing
// flow3 gather from disp_pred, back3 gather from back_disp_pred, superpixel
// labels + counts, fg scatter map. flow initialized to flow3.
__global__ void ml_prep_points_kernel(const float* __restrict__ src,
                                      const float* __restrict__ refpc,
                                      const float* __restrict__ dp,
                                      const float* __restrict__ bdp,
                                      const float* __restrict__ vsp,
                                      const float* __restrict__ extp,
                                      float* __restrict__ flow,
                                      float* __restrict__ flow3,
                                      float* __restrict__ back3,
                                      int*   __restrict__ labels,
                                      float* __restrict__ counts,
                                      float* __restrict__ fgmap) {
    int idx = blockIdx.x * blockDim.x + threadIdx.x;
    if (idx >= Bc * Nc) return;
    int b = idx / Nc;

    float vs   = vsp[0];
    int   off0 = (int)floorf(extp[0] / vs);   // extents[0][0]
    int   off1 = (int)floorf(extp[2] / vs);   // extents[1][0]

    float px = src[idx*3+0], py = src[idx*3+1], pz = src[idx*3+2];

    // source voxel index -> forward flow gather
    int i0 = (int)floorf(px / vs) - off0; i0 = min(max(i0, 0), Hc - 1);
    int i1 = (int)floorf(py / vs) - off1; i1 = min(max(i1, 0), Wc - 1);
    float f0 = dp[((b*2+0)*Hc + i0)*Wc + i1];
    float f1 = dp[((b*2+1)*Hc + i0)*Wc + i1];
    flow3[idx*3+0] = f0; flow3[idx*3+1] = f1; flow3[idx*3+2] = 0.0f;
    flow [idx*3+0] = f0; flow [idx*3+1] = f1; flow [idx*3+2] = 0.0f;
    fgmap[b*Hc*Wc + i0*Wc + i1] = 1.0f;   // benign race: all write 1.0

    // ref voxel index -> back flow gather (indexed later by nn_f)
    float rx = refpc[idx*3+0], ry = refpc[idx*3+1];
    int r0 = (int)floorf(rx / vs) - off0; r0 = min(max(r0, 0), Hc - 1);
    int r1 = (int)floorf(ry / vs) - off1; r1 = min(max(r1, 0), Wc - 1);
    back3[idx*3+0] = bdp[((b*2+0)*Hc + r0)*Wc + r1];
    back3[idx*3+1] = bdp[((b*2+1)*Hc + r0)*Wc + r1];
    back3[idx*3+2] = 0.0f;

    // superpixel hash label (int32 wraparound like jax)
    int c0 = (int)floorf(px / 4.0f);
    int c1 = (int)floorf(py / 4.0f);
    int c2 = (int)floorf(pz / 4.0f);
    int h = (int)((unsigned)c0 * 73856093u) ^ (int)((unsigned)c1 * 19349663u)
          ^ (int)((unsigned)c2 * 83492791u);
    int lab = h % SPc; if (lab < 0) lab += SPc;
    labels[idx] = lab;
    atomicAdd(&counts[b*SPc + lab], 1.0f);
}

// --------------------------------------------- per-cell map pass + reductions
// non_empty, num_fg, num_bg, bg |disp| sum (wave-reduced before atomics).
__global__ void ml_prep_map_kernel(const float* __restrict__ vox,
                                   const float* __restrict__ bev,
                                   const float* __restrict__ dp,
                                   const float* __restrict__ fgmap,
                                   float* __restrict__ scal) {
    int cell = blockIdx.x * blockDim.x + threadIdx.x;   // 0..H*W-1
    int b = blockIdx.y;
    int h = cell / Wc, w = cell % Wc;

    float zs = 0.0f;
    const float* vz = vox + (size_t)(((b*2 + 1)*Hc + h)*Wc + w) * Zc;
    for (int z = 0; z < Zc; ++z) zs += vz[z];
    float ne = (zs > 0.0f) ? 1.0f : 0.0f;

    float fgc  = fgmap[b*Hc*Wc + cell] * ne;
    float bgc  = bev[(b*Hc + h)*Wc + w] * ne;
    float babs = (fabsf(dp[((b*2+0)*Hc + h)*Wc + w]) +
                  fabsf(dp[((b*2+1)*Hc + h)*Wc + w])) * bgc;

    for (int off = 16; off > 0; off >>= 1) {
        fgc  += __shfl_down(fgc,  off);
        bgc  += __shfl_down(bgc,  off);
        babs += __shfl_down(babs, off);
    }
    if ((threadIdx.x & 31) == 0) {
        atomicAdd(&scal[b*8 + 0], fgc);
        atomicAdd(&scal[b*8 + 1], bgc);
        atomicAdd(&scal[b*8 + 2], babs);
    }
}

// ----------------------------------------------------- WMMA NN-match kernel
// Each wave owns a 16-row strip of warped source points (A: 16x4 f32, K=3
// padded) and sweeps all 8192 targets in 16-column tiles (B: 4x16 f32) staged
// through LDS. One V_WMMA_F32_16X16X4_F32 per tile yields the 16x16 dot
// block; per-lane running argmin over (||t||^2 - 2*dot), wave32 xor-shuffle
// final reduction; ||w||^2 is added only at the end (row-constant).
__global__ __launch_bounds__(256)
void ml_nn_match_kernel(const float* __restrict__ src,
                        const float* __restrict__ flow,
                        const float* __restrict__ tgt,
                        int*   __restrict__ nn_out,
                        float* __restrict__ dmin_out) {
    __shared__ float s_t[CHUNK * 3];           // 24 KB target stage

    const int b    = blockIdx.y;
    const int wave = threadIdx.x >> 5;
    const int lane = threadIdx.x & 31;
    const int half = lane >> 4;
    const int l16  = lane & 15;
    const int i0   = (blockIdx.x * 8 + wave) * 16;

    // A fragment: row M = i0 + l16. half0 lanes carry (K0,K1)=(x,y),
    // half1 lanes carry (K2,K3)=(z,0)  [ISA 16x4 f32 A layout].
    const int arow = i0 + l16;
    float wx = src[(b*Nc + arow)*3 + 0] + flow[(b*Nc + arow)*3 + 0];
    float wy = src[(b*Nc + arow)*3 + 1] + flow[(b*Nc + arow)*3 + 1];
    float wz = src[(b*Nc + arow)*3 + 2] + flow[(b*Nc + arow)*3 + 2];
    v2f a;
    a[0] = half ? wz : wx;
    a[1] = half ? 0.0f : wy;
    float wpart = half ? (wz*wz) : (wx*wx + wy*wy);
    float wsq   = wpart + __shfl_xor(wpart, 16);   // full ||w_M||^2, M = l16

    float best[8];
    int   bidx[8];
#pragma unroll
    for (int r = 0; r < 8; ++r) { best[r] = 3.402823466e38f; bidx[r] = 0; }

    for (int cb = 0; cb < Nc; cb += CHUNK) {
        __syncthreads();
        const float* gsrc = tgt + (size_t)(b*Nc + cb) * 3;
        if (cb + CHUNK < Nc)
            __builtin_prefetch(tgt + (size_t)(b*Nc + cb + CHUNK) * 3 +
                               threadIdx.x * 16, 0, 1);
        for (int t = threadIdx.x; t < CHUNK * 3; t += 256)
            s_t[t] = gsrc[t];
        __syncthreads();

        for (int tt = 0; tt < CHUNK; tt += 16) {
            int tj = tt + l16;
            float bx = s_t[tj*3 + 0];
            float by = s_t[tj*3 + 1];
            float bz = s_t[tj*3 + 2];
            float tsq = bx*bx + by*by + bz*bz;     // ||t_N||^2, N = l16
            v2f bb;
            bb[0] = half ? bz : bx;                 // B 4x16 f32 layout
            bb[1] = half ? 0.0f : by;

            v8f c = {};
            c = __builtin_amdgcn_wmma_f32_16x16x4_f32(
                    false, a, false, bb, (short)0, c, false, false);

            int j = cb + tt + l16;
#pragma unroll
            for (int r = 0; r < 8; ++r) {
                float val = tsq - 2.0f * c[r];     // d2 minus row constant
                if (val < best[r]) { best[r] = val; bidx[r] = j; }
            }
        }
    }

    // xor-shuffle argmin across the 16-lane half (offsets stay in-half).
#pragma unroll
    for (int r = 0; r < 8; ++r) {
        float v  = best[r];
        int   ix = bidx[r];
        for (int off = 1; off < 16; off <<= 1) {
            float ov = __shfl_xor(v,  off);
            int   oi = __shfl_xor(ix, off);
            if (ov < v || (ov == v && oi < ix)) { v = ov; ix = oi; }
        }
        // writer lane also owns wsq of its row: half0 lane r -> row i0+r,
        // half1 lane (r+8) -> row i0+8+r.
        int myrow = -1;
        if (half == 0 && l16 == r)     myrow = i0 + r;
        if (half == 1 && l16 == r + 8) myrow = i0 + 8 + r;
        if (myrow >= 0) {
            nn_out[b*Nc + myrow]   = ix;
            dmin_out[b*Nc + myrow] = sqrtf(fmaxf(v + wsq, 0.0f));
        }
    }
}

// ------------------------------------------------- weighted segment moments
__global__ void ml_segstats_kernel(const float* __restrict__ src,
                                   const float* __restrict__ tgt,
                                   const int*   __restrict__ nn,
                                   const float* __restrict__ dmin,
                                   const int*   __restrict__ labels,
                                   const int*   __restrict__ conf,
                                   float* __restrict__ seg) {
    int idx = blockIdx.x * blockDim.x + threadIdx.x;
    if (idx >= Bc * Nc) return;
    int b = idx / Nc;
    int k = labels[idx];
    float w = conf[0] ? __expf(-dmin[idx]) : 1.0f;   // THETA0 = 1

    float p0 = src[idx*3+0], p1 = src[idx*3+1], p2 = src[idx*3+2];
    int q = nn[idx];
    const float* qp = tgt + (size_t)(b*Nc + q) * 3;
    float q0 = qp[0], q1 = qp[1], q2 = qp[2];

    float* s = seg + (size_t)(b*SPc + k) * 16;
    atomicAdd(&s[0], w);
    atomicAdd(&s[1], w*p0); atomicAdd(&s[2], w*p1); atomicAdd(&s[3], w*p2);
    atomicAdd(&s[4], w*q0); atomicAdd(&s[5], w*q1); atomicAdd(&s[6], w*q2);
    atomicAdd(&s[7],  w*p0*q0); atomicAdd(&s[8],  w*p0*q1); atomicAdd(&s[9],  w*p0*q2);
    atomicAdd(&s[10], w*p1*q0); atomicAdd(&s[11], w*p1*q1); atomicAdd(&s[12], w*p1*q2);
    atomicAdd(&s[13], w*p2*q0); atomicAdd(&s[14], w*p2*q1); atomicAdd(&s[15], w*p2*q2);
}

// ------------------------------------------------------ per-segment Kabsch
// 3x3 SVD via cyclic Jacobi on H^T H; R = V diag(1,1,det(V)det(U)) U^T.
__global__ void ml_kabsch_kernel(const float* __restrict__ seg,
                                 const float* __restrict__ counts,
                                 float* __restrict__ Rk,
                                 float* __restrict__ trk,
                                 float* __restrict__ validk) {
    int idx = blockIdx.x * blockDim.x + threadIdx.x;
    if (idx >= Bc * SPc) return;

    const float* s = seg + (size_t)idx * 16;
    float sw = s[0] + 1e-8f;
    float mp[3], mq[3];
    for (int i = 0; i < 3; ++i) { mp[i] = s[1+i] / sw; mq[i] = s[4+i] / sw; }
    float H[3][3];
    for (int i = 0; i < 3; ++i)
        for (int j = 0; j < 3; ++j)
            H[i][j] = s[7 + 3*i + j] - sw * mp[i] * mq[j];

    bool valid = counts[idx] >= 3.0f;

    float R[3][3] = {{1,0,0},{0,1,0},{0,0,1}};
    float tr[3] = {0, 0, 0};

    if (valid) {
        // S = H^T H
        float S[3][3];
        for (int i = 0; i < 3; ++i)
            for (int j = 0; j < 3; ++j)
                S[i][j] = H[0][i]*H[0][j] + H[1][i]*H[1][j] + H[2][i]*H[2][j];
        float V[3][3] = {{1,0,0},{0,1,0},{0,0,1}};
        const int PP[3] = {0, 0, 1}, QQ[3] = {1, 2, 2};
        for (int sweep = 0; sweep < 10; ++sweep) {
            for (int e = 0; e < 3; ++e) {
                int p = PP[e], q = QQ[e];
                float apq = S[p][q];
                if (fabsf(apq) < 1e-14f) continue;
                float tau = (S[q][q] - S[p][p]) / (2.0f * apq);
                float t = ((tau >= 0.0f) ? 1.0f : -1.0f) /
                          (fabsf(tau) + sqrtf(1.0f + tau*tau));
                float cc = 1.0f / sqrtf(1.0f + t*t);
                float ss = t * cc;
                for (int k2 = 0; k2 < 3; ++k2) {
                    float a1 = S[k2][p], a2 = S[k2][q];
                    S[k2][p] = cc*a1 - ss*a2; S[k2][q] = ss*a1 + cc*a2;
                }
                for (int k2 = 0; k2 < 3; ++k2) {
                    float a1 = S[p][k2], a2 = S[q][k2];
                    S[p][k2] = cc*a1 - ss*a2; S[q][k2] = ss*a1 + cc*a2;
                }
                for (int k2 = 0; k2 < 3; ++k2) {
                    float a1 = V[k2][p], a2 = V[k2][q];
                    V[k2][p] = cc*a1 - ss*a2; V[k2][q] = ss*a1 + cc*a2;
                }
            }
        }
        // sort eigenvalues desc, permute V columns
        float ev[3] = {S[0][0], S[1][1], S[2][2]};
        int ord[3] = {0, 1, 2};
        for (int i = 0; i < 2; ++i)
            for (int j = i + 1; j < 3; ++j)
                if (ev[ord[j]] > ev[ord[i]]) { int t2 = ord[i]; ord[i] = ord[j]; ord[j] = t2; }
        float Vs[3][3];
        for (int i = 0; i < 3; ++i)
            for (int j = 0; j < 3; ++j) Vs[i][j] = V[i][ord[j]];

        // U columns = H v / sigma, fallback via cross products
        float U[3][3]; float nrm[3];
        for (int j = 0; j < 3; ++j) {
            float u0 = H[0][0]*Vs[0][j] + H[0][1]*Vs[1][j] + H[0][2]*Vs[2][j];
            float u1 = H[1][0]*Vs[0][j] + H[1][1]*Vs[1][j] + H[1][2]*Vs[2][j];
            float u2 = H[2][0]*Vs[0][j] + H[2][1]*Vs[1][j] + H[2][2]*Vs[2][j];
            float n = sqrtf(u0*u0 + u1*u1 + u2*u2);
            nrm[j] = n;
            float inv = 1.0f / fmaxf(n, 1e-12f);
            U[0][j] = u0*inv; U[1][j] = u1*inv; U[2][j] = u2*inv;
        }
        if (nrm[2] < 1e-8f) {
            float x = U[1][0]*U[2][1] - U[2][0]*U[1][1];
            float y = U[2][0]*U[0][1] - U[0][0]*U[2][1];
            float z = U[0][0]*U[1][1] - U[1][0]*U[0][1];
            float inv = 1.0f / fmaxf(sqrtf(x*x + y*y + z*z), 1e-12f);
            U[0][2] = x*inv; U[1][2] = y*inv; U[2][2] = z*inv;
        }
        float d = det3(Vs) * det3(U);
        // R = Vs * diag(1,1,d) * U^T
        for (int i = 0; i < 3; ++i)
            for (int j = 0; j < 3; ++j)
                R[i][j] = Vs[i][0]*U[j][0] + Vs[i][1]*U[j][1] + d*Vs[i][2]*U[j][2];
        for (int i = 0; i < 3; ++i)
            tr[i] = mq[i] - (R[i][0]*mp[0] + R[i][1]*mp[1] + R[i][2]*mp[2]);
    }

    float* Ro = Rk + (size_t)idx * 9;
    for (int i = 0; i < 3; ++i)
        for (int j = 0; j < 3; ++j) Ro[3*i + j] = R[i][j];
    float* to = trk + (size_t)idx * 3;
    to[0] = tr[0]; to[1] = tr[1]; to[2] = tr[2];
    validk[idx] = valid ? 1.0f : 0.0f;
}

// -------------------------------------------------------------- flow update
__global__ void ml_flow_update_kernel(const float* __restrict__ src,
                                      const int*   __restrict__ labels,
                                      const float* __restrict__ Rk,
                                      const float* __restrict__ trk,
                                      float* __restrict__ flow) {
    int idx = blockIdx.x * blockDim.x + threadIdx.x;
    if (idx >= Bc * Nc) return;
    int b = idx / Nc;
    int k = labels[idx];
    const float* R = Rk  + (size_t)(b*SPc + k) * 9;
    const float* t = trk + (size_t)(b*SPc + k) * 3;
    float p0 = src[idx*3+0], p1 = src[idx*3+1], p2 = src[idx*3+2];
    flow[idx*3+0] = R[0]*p0 + R[1]*p1 + R[2]*p2 + t[0] - p0;
    flow[idx*3+1] = R[3]*p0 + R[4]*p1 + R[5]*p2 + t[1] - p1;
    flow[idx*3+2] = R[6]*p0 + R[7]*p1 + R[8]*p2 + t[2] - p2;
}

// ------------------------------------------------------- final fg statistics
__global__ void ml_final_stats_kernel(const float* __restrict__ flow,
                                      const float* __restrict__ flow3,
                                      const float* __restrict__ back3,
                                      const float* __restrict__ dmin,
                                      const int*   __restrict__ nn,
                                      const int*   __restrict__ labels,
                                      const float* __restrict__ validk,
                                      const int*   __restrict__ conf,
                                      float* __restrict__ scal) {
    int idx = blockIdx.x * blockDim.x + threadIdx.x;
    if (idx >= Bc * Nc) return;
    int b = idx / Nc;

    float s0 = flow[idx*3+0], s1 = flow[idx*3+1], s2 = flow[idx*3+2];
    int nf = nn[idx];
    const float* bk = back3 + (size_t)(b*Nc + nf) * 3;
    float d0 = s0 + bk[0], d1 = s1 + bk[1], d2 = s2 + bk[2];
    float flow_diff = sqrtf(d0*d0 + d1*d1 + d2*d2);
    float pos_diff  = dmin[idx];

    float validity = validk[b*SPc + labels[idx]];
    if (conf[0])
        validity *= ((flow_diff < 3.0f) ? 1.0f : 0.0f) *
                    ((pos_diff  < 1.0f) ? 1.0f : 0.0f);

    float fg = fabsf(s0 - flow3[idx*3+0]) + fabsf(s1 - flow3[idx*3+1])
             + fabsf(s2 - flow3[idx*3+2]);

    float fsum = fg * validity, vsum = validity;
    for (int off = 16; off > 0; off >>= 1) {
        fsum += __shfl_down(fsum, off);
        vsum += __shfl_down(vsum, off);
    }
    if ((threadIdx.x & 31) == 0) {
        atomicAdd(&scal[b*8 + 3], fsum);
        atomicAdd(&scal[b*8 + 4], vsum);
    }
}

// ------------------------------------------------------------- loss assembly
__global__ void ml_final_loss_kernel(const float* __restrict__ scal,
                                     float* __restrict__ out) {
    __shared__ float ls[Bc];
    int b = threadIdx.x;
    if (b < Bc) {
        const float* s = scal + b*8;
        float num_fg = s[0], num_bg = s[1], bg_abs = s[2];
        float fg_num = s[3], fg_den = s[4];
        float fg_loss = fg_num / (fg_den + 1e-6f);
        float bg_loss = bg_abs / (num_bg + 1e-6f);
        ls[b] = (fg_loss * num_fg + 0.05f * bg_loss * num_bg) /
                (num_fg + num_bg + 1e-6f);
    }
    __syncthreads();
    if (threadIdx.x == 0) {
        float m = 0.0f;
        for (int i = 0; i < Bc; ++i) m += ls[i];
        out[0] = m / (float)Bc;
    }
}

// ---------------------------------------------------------------------------
extern "C" void kernel_launch(void* const* d_in, const int* in_sizes, int n_in,
                              void* d_out, int out_size, void* d_ws, size_t ws_size,
                              hipStream_t stream) {
    (void)in_sizes; (void)n_in; (void)out_size; (void)ws_size;

    const float* dp   = (const float*)d_in[0];
    const float* bdp  = (const float*)d_in[1];
    const float* src  = (const float*)d_in[2];
    const float* tgt  = (const float*)d_in[3];
    const float* refp = (const float*)d_in[4];
    const float* vsp  = (const float*)d_in[7];
    const float* extp = (const float*)d_in[8];
    const float* vox  = (const float*)d_in[9];
    const float* bev  = (const float*)d_in[10];
    const int*   conf = (const int*)d_in[11];
    float* out = (float*)d_out;

    // workspace layout (floats first, then ints)
    float* f = (float*)d_ws;
    size_t o = 0;
    float* flow   = f + o; o += (size_t)Bc*Nc*3;
    float* flow3  = f + o; o += (size_t)Bc*Nc*3;
    float* back3  = f + o; o += (size_t)Bc*Nc*3;
    float* dmin   = f + o; o += (size_t)Bc*Nc;
    float* counts = f + o; o += (size_t)Bc*SPc;
    float* seg    = f + o; o += (size_t)Bc*SPc*16;
    float* Rk     = f + o; o += (size_t)Bc*SPc*9;
    float* trk    = f + o; o += (size_t)Bc*SPc*3;
    float* validk = f + o; o += (size_t)Bc*SPc;
    float* fgmap  = f + o; o += (size_t)Bc*Hc*Wc;
    float* scal   = f + o; o += (size_t)Bc*8;
    int* ip = (int*)(f + o);
    int* labels = ip;
    int* nn     = ip + (size_t)Bc*Nc;

    const int TB = 256;
    const int ptBlocks = (Bc*Nc + TB - 1) / TB;

    ml_init_kernel<<<(Bc*Hc*Wc + TB - 1) / TB, TB, 0, stream>>>(fgmap, counts, scal);
    ml_prep_points_kernel<<<ptBlocks, TB, 0, stream>>>(
        src, refp, dp, bdp, vsp, extp, flow, flow3, back3, labels, counts, fgmap);
    ml_prep_map_kernel<<<dim3(Hc*Wc / TB, Bc), TB, 0, stream>>>(
        vox, bev, dp, fgmap, scal);

    for (int it = 0; it < ITERc; ++it) {
        ml_nn_match_kernel<<<dim3(Nc / (16 * 8), Bc), TB, 0, stream>>>(
            src, flow, tgt, nn, dmin);
        ml_zero_seg_kernel<<<4, TB, 0, stream>>>(seg);
        ml_segstats_kernel<<<ptBlocks, TB, 0, stream>>>(
            src, tgt, nn, dmin, labels, conf, seg);
        ml_kabsch_kernel<<<1, TB, 0, stream>>>(seg, counts, Rk, trk, validk);
        ml_flow_update_kernel<<<ptBlocks, TB, 0, stream>>>(
            src, labels, Rk, trk, flow);
    }

    // final NN match on pseudo flow
    ml_nn_match_kernel<<<dim3(Nc / (16 * 8), Bc), TB, 0, stream>>>(
        src, flow, tgt, nn, dmin);
    ml_final_stats_kernel<<<ptBlocks, TB, 0, stream>>>(
        flow, flow3, back3, dmin, nn, labels, validk, conf, scal);
    ml_final_loss_kernel<<<1, 32, 0, stream>>>(scal, out);
}